// FRRoiProcessLayer_29635274342938
// MI455X (gfx1250) — compile-verified
//
#include <hip/hip_runtime.h>
#include <cstdint>
#include <cstddef>

#define POOLED   7
#define FH       64
#define FW       64
#define FC       256
#define NROI     512
#define CG       8                 // channel-groups per ROI (blocks per ROI)
#define WAVES    4                 // wave32s per block
#define BLOCK    (WAVES * 32)      // 128 threads
#define CPB      (FC / CG)         // 32 channels per block
#define CPW      (CPB / WAVES)     // 8 channels per wave

// ---------------- CDNA5 async global->LDS copy helpers -----------------
#if defined(__AMDGCN__)
typedef int v4i __attribute__((ext_vector_type(4)));
typedef __attribute__((address_space(1))) v4i gv4i;   // global
typedef __attribute__((address_space(3))) v4i lv4i;   // LDS

#if __has_builtin(__builtin_amdgcn_global_load_async_to_lds_b128)
__device__ __forceinline__ void async_copy_b128(const float* src, float* dst_lds) {
    __builtin_amdgcn_global_load_async_to_lds_b128((gv4i*)src, (lv4i*)dst_lds, 0, 0);
}
#else
__device__ __forceinline__ void async_copy_b128(const float* src, float* dst_lds) {
    unsigned lds_off = (unsigned)(uintptr_t)dst_lds;   // low 32 bits = LDS byte address
    asm volatile("global_load_async_to_lds_b128 %0, %1, off"
                 :: "v"(lds_off), "v"(src) : "memory");
}
#endif

#if __has_builtin(__builtin_amdgcn_s_wait_asynccnt)
__device__ __forceinline__ void wait_async0() {
    __builtin_amdgcn_s_wait_asynccnt(0);
    asm volatile("" ::: "memory");
}
#else
__device__ __forceinline__ void wait_async0() {
    asm volatile("s_wait_asynccnt 0" ::: "memory");
}
#endif
#else
// host pass: never codegen'd
__device__ __forceinline__ void async_copy_b128(const float*, float*) {}
__device__ __forceinline__ void wait_async0() {}
#endif
// -----------------------------------------------------------------------

__global__ __launch_bounds__(BLOCK)
void roi_pool_kernel(const float* __restrict__ features,
                     const float* __restrict__ rois,
                     float* __restrict__ out)
{
    __shared__ float smem[WAVES * FH * FW];   // 4 x 16KB = 64KB, wave-private slices

    const int blk  = blockIdx.x;
    const int n    = blk / CG;
    const int cg   = blk % CG;
    const int wave = threadIdx.x >> 5;
    const int lane = threadIdx.x & 31;

    // ---- ROI parameters (uniform across block; matches jnp.round = rint) ----
    const float* r = rois + n * 5;
    const int   b  = (int)r[0];
    const float sc = 1.0f / 16.0f;
    const int x1 = (int)rintf(r[1] * sc);
    const int y1 = (int)rintf(r[2] * sc);
    const int x2 = (int)rintf(r[3] * sc);
    const int y2 = (int)rintf(r[4] * sc);
    const float roi_w = (float)max(x2 - x1 + 1, 1);
    const float roi_h = (float)max(y2 - y1 + 1, 1);
    const float bsw = roi_w / (float)POOLED;
    const float bsh = roi_h / (float)POOLED;

    // Staged window: rows [r0,r1) and 16B-aligned cols [c0,c1) cover every bin's range.
    const int r0 = min(max(y1, 0), FH);
    const int r1 = min(max((int)ceilf(7.0f * bsh) + y1, 0), FH);
    const int nrows = r1 - r0;                       // 0..64
    const int c0 = (min(max(x1, 0), FW)) & ~3;
    int c1 = (min(max((int)ceilf(7.0f * bsw) + x1, 0), FW) + 3) & ~3;  // <= 64 (FW mult of 4)
    const int cpr = (c1 - c0) >> 2;                  // 16B chunks per row (0..16)
    const int nchunks = nrows * cpr;

    float* wbuf = smem + wave * (FH * FW);

    const int cbase = cg * CPB + wave * CPW;
    for (int i = 0; i < CPW; ++i) {
        const int c = cbase + i;
        const float* gbase = features + ((size_t)(b * FC + c)) * (FH * FW);

        // ---- async stage ROI window of this channel into wave-private LDS ----
        for (int k = lane; k < nchunks; k += 32) {
            const int row = k / cpr;
            const int col = c0 + ((k - row * cpr) << 2);
            async_copy_b128(gbase + (size_t)(r0 + row) * FW + col,
                            wbuf + row * FW + col);
        }
        wait_async0();

        // ---- 49 bins, lanes 0..31 take bins {lane, lane+32} ----
        #pragma unroll
        for (int pass = 0; pass < 2; ++pass) {
            const int bin = lane + pass * 32;
            if (bin < POOLED * POOLED) {
                const int ph = bin / POOLED;
                const int pw = bin - ph * POOLED;
                const int hs = min(max((int)floorf((float)ph * bsh) + y1, 0), FH);
                const int he = min(max((int)ceilf((float)(ph + 1) * bsh) + y1, 0), FH);
                const int ws = min(max((int)floorf((float)pw * bsw) + x1, 0), FW);
                const int we = min(max((int)ceilf((float)(pw + 1) * bsw) + x1, 0), FW);
                float m = 0.0f;                       // empty bin -> 0 (matches ref)
                if (he > hs && we > ws) {
                    m = -__builtin_huge_valf();
                    for (int h = hs; h < he; ++h) {
                        const float* rowp = wbuf + (h - r0) * FW;
                        for (int w = ws; w < we; ++w)
                            m = fmaxf(m, rowp[w]);
                    }
                }
                out[(((size_t)n * FC + c) * (POOLED * POOLED)) + ph * POOLED + pw] = m;
            }
        }
        // wbuf safely reused next channel: all ds reads are consumed (DScnt==0)
        // before the out-store that precedes the next async issues.
    }
}

extern "C" void kernel_launch(void* const* d_in, const int* in_sizes, int n_in,
                              void* d_out, int out_size, void* d_ws, size_t ws_size,
                              hipStream_t stream) {
    (void)in_sizes; (void)n_in; (void)out_size; (void)d_ws; (void)ws_size;
    const float* features = (const float*)d_in[0];   // (2,256,64,64) f32
    const float* rois     = (const float*)d_in[1];   // (512,5) f32
    float* out            = (float*)d_out;           // (512,256,7,7) f32

    dim3 grid(NROI * CG);
    dim3 block(BLOCK);
    roi_pool_kernel<<<grid, block, 0, stream>>>(features, rois, out);
}